// AttentionLayer_23244363006589
// MI455X (gfx1250) — compile-verified
//
#include <hip/hip_runtime.h>
#include <math.h>

// Problem constants (from reference setup_inputs)
#define EDIM 512
#define KDIM 64
#define NROWS 2048   // B*L = 32*64
#define EPSF 1e-5f

typedef __attribute__((ext_vector_type(2))) float v2f;
typedef __attribute__((ext_vector_type(8))) float v8f;

// Workspace layout (floats): [0..511] gvy = gamma*(W1 @ w2y), [512] G = sum(gvy),
//                            [1024 .. 1024 + NROWS*EDIM) z buffer

// --- Kernel 1: gvy[e] = gamma[e] * sum_f W1[e,f] * W2[E+f] -------------------
__global__ void k_vy(const float* __restrict__ W1, const float* __restrict__ W2,
                     const float* __restrict__ gamma, float* __restrict__ ws) {
  int e = blockIdx.x * blockDim.x + threadIdx.x;
  if (e >= EDIM) return;
  const float* w2y = W2 + EDIM;
  const float* row = W1 + (size_t)e * EDIM;
  float acc = 0.f;
  for (int f = 0; f < EDIM; ++f) acc += row[f] * w2y[f];
  ws[e] = acc * gamma[e];
}

// --- Kernel 1b: G = sum(gvy) (deterministic tree reduce) ---------------------
__global__ void k_gsum(float* __restrict__ ws) {
  __shared__ float red[256];
  int t = threadIdx.x;
  red[t] = ws[t] + ws[t + 256];
  __syncthreads();
  for (int s = 128; s > 0; s >>= 1) {
    if (t < s) red[t] += red[t + s];
    __syncthreads();
  }
  if (t == 0) ws[512] = red[0];
}

// --- Kernel 2: per-n fused LN-stats + softmax + weighted combine -------------
// One workgroup (256 threads = 8 waves) per n. y[n] (128KB) staged into LDS
// exactly once (CDNA5: 320KB LDS per WGP); stats computed during staging.
__global__ void __launch_bounds__(256) k_attn(const float* __restrict__ y,
                                              const float* __restrict__ gamma,
                                              const float* __restrict__ beta,
                                              const float* __restrict__ ws,
                                              float* __restrict__ z) {
  __shared__ float sy[KDIM * EDIM];   // 128 KB
  __shared__ float s_gvy[EDIM];
  __shared__ float s_m[KDIM], s_rs[KDIM], s_r[KDIM], s_w[KDIM];
  __shared__ float s_c;

  const int n = blockIdx.x;
  const int t = threadIdx.x;
  const int wv = t >> 5;
  const int lane = t & 31;

  s_gvy[t] = ws[t];
  s_gvy[t + 256] = ws[t + 256];
  const float G = ws[512];
  __syncthreads();

  // Stage 8 rows per wave; accumulate Sum(y), Sum(y^2), Sum(y*gvy) in flight.
  const float* ybase = y + (size_t)n * (KDIM * EDIM);
  for (int i = 0; i < 8; ++i) {
    const int k = wv * 8 + i;
    const float* row = ybase + k * EDIM;
    float s1 = 0.f, s2 = 0.f, s3 = 0.f;
#pragma unroll
    for (int c = 0; c < 4; ++c) {
      const int e = c * 128 + lane * 4;
      float4 v = *(const float4*)(row + e);
      *(float4*)(&sy[k * EDIM + e]) = v;
      s1 += v.x + v.y + v.z + v.w;
      s2 += v.x * v.x + v.y * v.y + v.z * v.z + v.w * v.w;
      s3 += v.x * s_gvy[e] + v.y * s_gvy[e + 1] + v.z * s_gvy[e + 2] + v.w * s_gvy[e + 3];
    }
#pragma unroll
    for (int m = 16; m >= 1; m >>= 1) {
      s1 += __shfl_xor(s1, m, 32);
      s2 += __shfl_xor(s2, m, 32);
      s3 += __shfl_xor(s3, m, 32);
    }
    if (lane == 0) {
      const float mean = s1 * (1.f / EDIM);
      const float var = s2 * (1.f / EDIM) - mean * mean;
      const float rs = rsqrtf(var + EPSF);
      s_m[k] = mean;
      s_rs[k] = rs;
      s_r[k] = rs * (s3 - mean * G);   // logit up to per-n constant (cancels in softmax)
    }
  }
  __syncthreads();

  // Softmax over K=64 by wave 0; w_k = prob_k * rs_k ; c = sum_k w_k * mean_k
  if (wv == 0) {
    float r0 = s_r[lane], r1 = s_r[lane + 32];
    float mx = fmaxf(r0, r1);
#pragma unroll
    for (int m = 16; m >= 1; m >>= 1) mx = fmaxf(mx, __shfl_xor(mx, m, 32));
    float e0 = expf(r0 - mx), e1 = expf(r1 - mx);
    float sum = e0 + e1;
#pragma unroll
    for (int m = 16; m >= 1; m >>= 1) sum += __shfl_xor(sum, m, 32);
    const float inv = 1.f / sum;
    const float w0 = e0 * inv * s_rs[lane];
    const float w1 = e1 * inv * s_rs[lane + 32];
    s_w[lane] = w0;
    s_w[lane + 32] = w1;
    float cp = w0 * s_m[lane] + w1 * s_m[lane + 32];
#pragma unroll
    for (int m = 16; m >= 1; m >>= 1) cp += __shfl_xor(cp, m, 32);
    if (lane == 0) s_c = cp;
  }
  __syncthreads();

  // z[n,e] = gamma[e]*(sum_k w_k*y[n,k,e] - c) + beta[e]  (from LDS)
  const float cc = s_c;
  float acc0 = 0.f, acc1 = 0.f;
#pragma unroll 4
  for (int k = 0; k < KDIM; ++k) {
    const float wk = s_w[k];
    acc0 += wk * sy[k * EDIM + t];
    acc1 += wk * sy[k * EDIM + t + 256];
  }
  z[(size_t)n * EDIM + t]       = gamma[t]       * (acc0 - cc) + beta[t];
  z[(size_t)n * EDIM + t + 256] = gamma[t + 256] * (acc1 - cc) + beta[t + 256];
}

// --- Kernel 3: out = gelu(x + z @ W1 + b1) via V_WMMA_F32_16X16X4_F32 --------
// One wave per 16x16 output tile; K-loop of 512 in steps of 4.
// A (16x4 f32): lanes0-15 M=lane {K=kb+0,kb+1}, lanes16-31 M=lane-16 {K=kb+2,kb+3}
// B (4x16 f32): mirrored with N in lanes.
__global__ void __launch_bounds__(256) k_gemm(const float* __restrict__ z,
                                              const float* __restrict__ W1,
                                              const float* __restrict__ b1,
                                              const float* __restrict__ x,
                                              float* __restrict__ out) {
  const int t = threadIdx.x;
  const int wv = t >> 5;
  const int lane = t & 31;
  const int tile = blockIdx.x * 8 + wv;   // 4096 tiles total
  const int mtile = tile >> 5;            // 512/16 = 32 n-tiles per m-row
  const int ntile = tile & 31;
  const int l15 = lane & 15;
  const int half = lane >> 4;
  const int mbase = mtile * 16;
  const int nbase = ntile * 16;
  const int col = nbase + l15;

  v8f acc = {};
  const float* arow = z + (size_t)(mbase + l15) * EDIM;
  for (int kb = 0; kb < EDIM; kb += 4) {
    const int ka = kb + half * 2;
    v2f a, b;
    a.x = arow[ka];
    a.y = arow[ka + 1];
    b.x = W1[(size_t)ka * EDIM + col];
    b.y = W1[(size_t)(ka + 1) * EDIM + col];
    acc = __builtin_amdgcn_wmma_f32_16x16x4_f32(
        /*neg_a=*/false, a, /*neg_b=*/false, b,
        /*c_mod=*/(short)0, acc, /*reuse_a=*/false, /*reuse_b=*/false);
  }

  const float bb = b1[col];
#pragma unroll
  for (int r = 0; r < 8; ++r) {
    const int m = mbase + r + half * 8;
    const float av = acc[r] + bb;
    const float tt = x[(size_t)m * EDIM + col] + av;
    // tanh-approx GELU (jax.nn.gelu default approximate=True)
    const float g = 0.5f * tt * (1.f + tanhf(0.7978845608f * (tt + 0.044715f * tt * tt * tt)));
    out[(size_t)m * EDIM + col] = g;
  }
}

extern "C" void kernel_launch(void* const* d_in, const int* in_sizes, int n_in,
                              void* d_out, int out_size, void* d_ws, size_t ws_size,
                              hipStream_t stream) {
  const float* x     = (const float*)d_in[0];
  const float* y     = (const float*)d_in[1];
  const float* gamma = (const float*)d_in[2];
  const float* beta  = (const float*)d_in[3];
  const float* W1    = (const float*)d_in[4];
  const float* b1    = (const float*)d_in[5];
  const float* W2    = (const float*)d_in[6];
  // d_in[7] = b2 (uniform logit shift: cancels in softmax), d_in[8] = K (fixed 64)
  float* ws   = (float*)d_ws;
  float* zbuf = ws + 1024;
  float* out  = (float*)d_out;

  k_vy  <<<2,     256, 0, stream>>>(W1, W2, gamma, ws);
  k_gsum<<<1,     256, 0, stream>>>(ws);
  k_attn<<<NROWS, 256, 0, stream>>>(y, gamma, beta, ws, zbuf);
  k_gemm<<<512,   256, 0, stream>>>(zbuf, W1, b1, x, out);
}